// GraphSAGE_27633819583002
// MI455X (gfx1250) — compile-verified
//
#include <hip/hip_runtime.h>
#include <hip/hip_bf16.h>

typedef __attribute__((ext_vector_type(2))) float v2f;
typedef __attribute__((ext_vector_type(8))) float v8f;

#define D 64
#define WPITCH 66   // LDS row pitch for W matrices (bank-conflict padding)

// ---------------------------------------------------------------------------
// Edge scatter: one wave (32 lanes) per edge, each lane handles float2 (2 feats)
// agg[dst] += feat[src]; optionally cnt[dst] += 1 (lane 0)
// ---------------------------------------------------------------------------
__global__ __launch_bounds__(256) void sage_scatter_kernel(
    const float* __restrict__ feat,
    const int* __restrict__ src,
    const int* __restrict__ dst,
    float* __restrict__ agg,
    float* __restrict__ cnt,   // nullptr -> skip count accumulation
    int nEdges)
{
    int e = blockIdx.x * 8 + (threadIdx.x >> 5);
    if (e >= nEdges) return;
    int lane = threadIdx.x & 31;

    int s = src[e];
    int d = dst[e];

    const float2* fp = (const float2*)(feat + (size_t)s * D);
    float2 v = fp[lane];

    float* ap = agg + (size_t)d * D + lane * 2;
    atomicAdd(ap,     v.x);
    atomicAdd(ap + 1, v.y);

    if (cnt != nullptr && lane == 0) {
        atomicAdd(cnt + d, 1.0f);
    }
}

// ---------------------------------------------------------------------------
// SAGE dense layer via V_WMMA_F32_16X16X4_F32:
//   out = (agg / max(cnt,1)) @ Wl^T + xin @ Wr^T + bias   [+ ReLU]
// One wave computes one 16-row x 64-col output tile (4 accumulator tiles).
// 8 waves (256 threads) per block -> 8 row-tiles per block.
// ---------------------------------------------------------------------------
__global__ __launch_bounds__(256) void sage_dense_kernel(
    const float* __restrict__ agg,
    const float* __restrict__ cnt,
    const float* __restrict__ xin,
    const float* __restrict__ Wl,
    const float* __restrict__ Wr,
    const float* __restrict__ bias,
    float* __restrict__ out,
    int nTiles,
    int doRelu)
{
    __shared__ float sWl[D * WPITCH];
    __shared__ float sWr[D * WPITCH];
    __shared__ float sB[D];

    // Stage W matrices + bias into LDS (coalesced)
    for (int i = threadIdx.x; i < D * D; i += 256) {
        int r = i >> 6;
        int c = i & 63;
        sWl[r * WPITCH + c] = Wl[i];
        sWr[r * WPITCH + c] = Wr[i];
    }
    if (threadIdx.x < D) sB[threadIdx.x] = bias[threadIdx.x];
    __syncthreads();

    int wave = threadIdx.x >> 5;
    int lane = threadIdx.x & 31;
    int tile = blockIdx.x * 8 + wave;
    if (tile >= nTiles) return;          // wave-uniform: EXEC stays all-ones

    int col   = lane & 15;               // N within 16-tile / M row within A
    int khalf = (lane >> 4) << 1;        // 0 or 2: K sub-pair selector

    int rowA = (tile << 4) + col;        // A-fragment row for this lane
    float inv = 1.0f / fmaxf(cnt[rowA], 1.0f);

    const float* aggRow = agg + (size_t)rowA * D;
    const float* xRow   = xin + (size_t)rowA * D;

    // Init accumulators with bias (broadcast along M)
    v8f acc[4];
#pragma unroll
    for (int nt = 0; nt < 4; ++nt) {
        float bv = sB[nt * 16 + col];
        acc[nt] = (v8f){bv, bv, bv, bv, bv, bv, bv, bv};
    }

#pragma unroll
    for (int k = 0; k < D; k += 4) {
        int kk = k + khalf;

        // A fragments: mean (agg*inv) and raw x
        v2f aM, aX;
        aM.x = aggRow[kk]     * inv;
        aM.y = aggRow[kk + 1] * inv;
        aX.x = xRow[kk];
        aX.y = xRow[kk + 1];

#pragma unroll
        for (int nt = 0; nt < 4; ++nt) {
            int wc = nt * 16 + col;      // output column = W row (B = W^T)
            v2f bL, bR;
            bL.x = sWl[wc * WPITCH + kk];
            bL.y = sWl[wc * WPITCH + kk + 1];
            bR.x = sWr[wc * WPITCH + kk];
            bR.y = sWr[wc * WPITCH + kk + 1];

            acc[nt] = __builtin_amdgcn_wmma_f32_16x16x4_f32(
                false, aM, false, bL, (short)0, acc[nt], false, false);
            acc[nt] = __builtin_amdgcn_wmma_f32_16x16x4_f32(
                false, aX, false, bR, (short)0, acc[nt], false, false);
        }
    }

    // Store: VGPR r -> row (tile*16 + r + 8*(lane>>4)), col nt*16 + (lane&15)
    int mbase = (tile << 4) + ((lane >> 4) << 3);
#pragma unroll
    for (int nt = 0; nt < 4; ++nt) {
#pragma unroll
        for (int r = 0; r < 8; ++r) {
            float v = acc[nt][r];
            if (doRelu) v = fmaxf(v, 0.0f);
            out[(size_t)(mbase + r) * D + nt * 16 + col] = v;
        }
    }
}

// ---------------------------------------------------------------------------
extern "C" void kernel_launch(void* const* d_in, const int* in_sizes, int n_in,
                              void* d_out, int out_size, void* d_ws, size_t ws_size,
                              hipStream_t stream) {
    const float* x   = (const float*)d_in[0];
    const int* eidx  = (const int*)d_in[1];
    const float* W1l = (const float*)d_in[2];
    const float* W1r = (const float*)d_in[3];
    const float* b1  = (const float*)d_in[4];
    const float* W2l = (const float*)d_in[5];
    const float* W2r = (const float*)d_in[6];
    const float* b2  = (const float*)d_in[7];
    float* out = (float*)d_out;

    int nNodes = in_sizes[0] / D;       // 100000
    int nEdges = in_sizes[1] / 2;       // 1600000
    const int* src = eidx;
    const int* dst = eidx + nEdges;

    // Workspace layout: [agg: N*D] [h: N*D] [cnt: N]
    float* agg = (float*)d_ws;
    float* h   = agg + (size_t)nNodes * D;
    float* cnt = h + (size_t)nNodes * D;

    int scatterBlocks = (nEdges + 7) / 8;
    int nTiles = nNodes / 16;           // 6250 (N divisible by 16)
    int denseBlocks = (nTiles + 7) / 8;

    // ---- Layer 1 ----
    hipMemsetAsync(agg, 0, (size_t)nNodes * D * sizeof(float), stream);
    hipMemsetAsync(cnt, 0, (size_t)nNodes * sizeof(float), stream);

    sage_scatter_kernel<<<scatterBlocks, 256, 0, stream>>>(
        x, src, dst, agg, cnt, nEdges);

    sage_dense_kernel<<<denseBlocks, 256, 0, stream>>>(
        agg, cnt, x, W1l, W1r, b1, h, nTiles, /*doRelu=*/1);

    // ---- Layer 2 (counts identical; reuse cnt, re-zero agg) ----
    hipMemsetAsync(agg, 0, (size_t)nNodes * D * sizeof(float), stream);

    sage_scatter_kernel<<<scatterBlocks, 256, 0, stream>>>(
        h, src, dst, agg, nullptr, nEdges);

    sage_dense_kernel<<<denseBlocks, 256, 0, stream>>>(
        agg, cnt, h, W2l, W2r, b2, out, nTiles, /*doRelu=*/0);
}